// myGATConv_edge_only_diff_13975823582063
// MI455X (gfx1250) — compile-verified
//
#include <hip/hip_runtime.h>
#include <hip/hip_bf16.h>

#define DFEAT   64
#define ALPHA   0.1f
#define ORDER   5
#define CHUNK   256   // edges staged per TDM chunk (fits tile_dim0 16-bit field)

#if __has_builtin(__builtin_amdgcn_tensor_load_to_lds)
#define USE_TDM 1
#else
#define USE_TDM 0
#endif

// ---------------------------------------------------------------- utilities
typedef unsigned int u32x4 __attribute__((ext_vector_type(4)));
typedef int          i32x8 __attribute__((ext_vector_type(8)));
typedef int          i32x4 __attribute__((ext_vector_type(4)));

#if USE_TDM
typedef __attribute__((address_space(3))) void lds_void_t;

__device__ __forceinline__ unsigned lds_byte_off(void* p) {
  // ptrtoint of an addrspace(3) pointer yields the LDS byte offset.
  return (unsigned)(unsigned long long)(lds_void_t*)p;
}

// 1-D DMA of `cnt` dwords from global -> LDS via the Tensor Data Mover.
// Descriptor bit layout per CDNA5 ISA §8 (groups 0/1; groups 2/3 zero => <=2D).
__device__ __forceinline__ void tdm_load_dwords(const void* gsrc, unsigned ldsoff,
                                                unsigned cnt) {
  unsigned long long ga = (unsigned long long)gsrc;
  u32x4 g0;
  g0[0] = 1u;                                   // count=1, user descriptor
  g0[1] = ldsoff;                               // lds_addr [63:32]
  g0[2] = (unsigned)ga;                         // global_addr[31:0]
  g0[3] = ((unsigned)(ga >> 32) & 0x01FFFFFFu)  // global_addr[56:32]
          | 0x80000000u;                        // type=2 ("image")
  i32x8 g1;
  g1[0] = (int)(2u << 16);                      // data_size=2 (4 bytes); wg_mask=0
  g1[1] = (int)((cnt & 0xFFFFu) << 16);         // tensor_dim0[15:0]
  g1[2] = (int)(((cnt >> 16) & 0xFFFFu)         // tensor_dim0[31:16]
          | (1u << 16));                        // tensor_dim1 = 1
  g1[3] = (int)((cnt & 0xFFFFu) << 16);         // tile_dim0 = cnt
  g1[4] = 1;                                    // tile_dim1 = 1, tile_dim2 = 0
  g1[5] = (int)cnt;                             // tensor_dim0_stride (low 32)
  g1[6] = 0;
  g1[7] = 0;
  i32x4 gz4 = {0, 0, 0, 0};
  i32x8 gz8 = {0, 0, 0, 0, 0, 0, 0, 0};
  // clang-23 / therock-10.0 form: 6 args (groups 0..3, extra group, cpol)
  __builtin_amdgcn_tensor_load_to_lds(g0, g1, gz4, gz4, gz8, 0);
}
#endif

// ---------------------------------------------------------------- kernels

// per-node init: softmax denom, etype mask, attn_sum scalar
__global__ void k_init_nodes(float* __restrict__ ssum, unsigned* __restrict__ mask,
                             float* __restrict__ attn, int N) {
  int i = blockIdx.x * blockDim.x + threadIdx.x;
  if (i < N) { ssum[i] = 0.0f; mask[i] = 0u; attn[i] = ALPHA; }
}

// x = ALPHA * feat   (vectorized float4 over ND/4)
__global__ void k_init_x(float* __restrict__ x, const float* __restrict__ feat, int nd4) {
  int i = blockIdx.x * blockDim.x + threadIdx.x;
  if (i < nd4) {
    float4 f = ((const float4*)feat)[i];
    f.x *= ALPHA; f.y *= ALPHA; f.z *= ALPHA; f.w *= ALPHA;
    ((float4*)x)[i] = f;
  }
}

// per-edge: record which edge types hit each dst (exact segment_max prep)
__global__ void k_edge_mask(const int* __restrict__ ef, const int* __restrict__ dst,
                            unsigned* __restrict__ mask, int E) {
  int e = blockIdx.x * blockDim.x + threadIdx.x;
  if (e < E) atomicOr(&mask[dst[e]], 1u << (ef[e] & 31));
}

// per-node: m = max over present etypes of edge_emb[t]
__global__ void k_node_max(const unsigned* __restrict__ mask,
                           const float* __restrict__ emb,
                           float* __restrict__ mmax, int N, int NT) {
  int i = blockIdx.x * blockDim.x + threadIdx.x;
  if (i >= N) return;
  unsigned m = mask[i];
  float v = -3.0e38f;
  for (int t = 0; t < NT; ++t)
    if ((m >> t) & 1u) v = fmaxf(v, emb[t]);
  mmax[i] = v;
}

// per-edge: ex = exp(e - m[dst]); stash in a[]; accumulate denom
__global__ void k_edge_exp(const int* __restrict__ ef, const int* __restrict__ dst,
                           const float* __restrict__ emb, const float* __restrict__ mmax,
                           float* __restrict__ a, float* __restrict__ ssum, int E) {
  int e = blockIdx.x * blockDim.x + threadIdx.x;
  if (e >= E) return;
  int d = dst[e];
  float ex = __expf(emb[ef[e]] - mmax[d]);
  a[e] = ex;
  atomicAdd(&ssum[d], ex);
}

// per-edge: a = ex / denom[dst]
__global__ void k_edge_norm(const int* __restrict__ dst, const float* __restrict__ ssum,
                            float* __restrict__ a, int E) {
  int e = blockIdx.x * blockDim.x + threadIdx.x;
  if (e < E) a[e] = a[e] / ssum[dst[e]];
}

// zero the aggregation buffers
__global__ void k_zero(float* __restrict__ aggft, float* __restrict__ aggas,
                       int nd4, int N) {
  int i = blockIdx.x * blockDim.x + threadIdx.x;
  if (i < nd4) ((float4*)aggft)[i] = make_float4(0.f, 0.f, 0.f, 0.f);
  if (i < N) aggas[i] = 0.0f;
}

// edge aggregation: one edge per wave32; lane l covers features 2l, 2l+1.
// Edge metadata (src, dst, a) is staged into LDS by the Tensor Data Mover,
// double-buffered across chunks; feature gathers hit L2 (x is 12.8MB << 192MB).
__global__ void k_edge_agg(const int* __restrict__ src, const int* __restrict__ dst,
                           const float* __restrict__ a, const float* __restrict__ x,
                           const float* __restrict__ attn,
                           float* __restrict__ aggft, float* __restrict__ aggas, int E) {
  __shared__ int   s_src[2][CHUNK];
  __shared__ int   s_dst[2][CHUNK];
  __shared__ float s_a[2][CHUNK];
  const int wave = threadIdx.x >> 5;
  const int lane = threadIdx.x & 31;
  const int nChunks = (E + CHUNK - 1) / CHUNK;
#if USE_TDM
  const bool issuer = (threadIdx.x < 32);   // wave 0 drives the TDM
  if (issuer) {
    int c0 = blockIdx.x;
    if (c0 < nChunks) {
      int base = c0 * CHUNK;
      int cnt  = min(CHUNK, E - base);
      tdm_load_dwords(src + base, lds_byte_off(&s_src[0][0]), (unsigned)cnt);
      tdm_load_dwords(dst + base, lds_byte_off(&s_dst[0][0]), (unsigned)cnt);
      tdm_load_dwords(a   + base, lds_byte_off(&s_a[0][0]),   (unsigned)cnt);
    }
  }
  int buf = 0;
  for (int c = blockIdx.x; c < nChunks; c += (int)gridDim.x) {
    int cn = c + (int)gridDim.x;
    if (issuer) {
      if (cn < nChunks) {       // prefetch next chunk into the other buffer
        int nb = buf ^ 1;
        int base = cn * CHUNK;
        int cnt  = min(CHUNK, E - base);
        tdm_load_dwords(src + base, lds_byte_off(&s_src[nb][0]), (unsigned)cnt);
        tdm_load_dwords(dst + base, lds_byte_off(&s_dst[nb][0]), (unsigned)cnt);
        tdm_load_dwords(a   + base, lds_byte_off(&s_a[nb][0]),   (unsigned)cnt);
        __builtin_amdgcn_s_wait_tensorcnt(3);   // current chunk's 3 DMAs done
      } else {
        __builtin_amdgcn_s_wait_tensorcnt(0);
      }
    }
    __syncthreads();
    const int base = c * CHUNK;
    const int cnt  = min(CHUNK, E - base);
    for (int i = wave; i < cnt; i += 8) {
      int   s = s_src[buf][i];
      int   d = s_dst[buf][i];
      float w = s_a[buf][i];
      const float2 xv = *(const float2*)(x + ((size_t)s << 6) + (lane << 1));
      float* o = aggft + ((size_t)d << 6) + (lane << 1);
      atomicAdd(o,     w * xv.x);
      atomicAdd(o + 1, w * xv.y);
      if (lane == 0) atomicAdd(&aggas[d], w * attn[s]);   // attn_sum is per-node scalar
    }
    __syncthreads();   // all reads of buf done before it is DMA-overwritten
    buf ^= 1;
  }
#else
  const int wpb = blockDim.x >> 5;
  for (int e = blockIdx.x * wpb + wave; e < E; e += (int)gridDim.x * wpb) {
    int   s = src[e];
    int   d = dst[e];
    float w = a[e];
    const float2 xv = *(const float2*)(x + ((size_t)s << 6) + (lane << 1));
    float* o = aggft + ((size_t)d << 6) + (lane << 1);
    atomicAdd(o,     w * xv.x);
    atomicAdd(o + 1, w * xv.y);
    if (lane == 0) atomicAdd(&aggas[d], w * attn[s]);
  }
#endif
}

// x = 0.9*aggft + 0.1*feat ; attn = 0.9*aggas + 0.01
__global__ void k_node_update(const float* __restrict__ feat,
                              const float* __restrict__ aggft,
                              const float* __restrict__ aggas,
                              float* __restrict__ x, float* __restrict__ attn,
                              int nd4, int N) {
  int i = blockIdx.x * blockDim.x + threadIdx.x;
  if (i < nd4) {
    float4 g = ((const float4*)aggft)[i];
    float4 f = ((const float4*)feat)[i];
    float4 r;
    r.x = (1.0f - ALPHA) * g.x + ALPHA * f.x;
    r.y = (1.0f - ALPHA) * g.y + ALPHA * f.y;
    r.z = (1.0f - ALPHA) * g.z + ALPHA * f.z;
    r.w = (1.0f - ALPHA) * g.w + ALPHA * f.w;
    ((float4*)x)[i] = r;
  }
  if (i < N) attn[i] = (1.0f - ALPHA) * aggas[i] + ALPHA * ALPHA;
}

// rst = x / attn[node]   (in place on d_out)
__global__ void k_finalize(float* __restrict__ x, const float* __restrict__ attn, int nd4) {
  int i = blockIdx.x * blockDim.x + threadIdx.x;
  if (i >= nd4) return;
  float inv = 1.0f / attn[i >> 4];       // 16 float4s per 64-wide node row
  float4 v = ((float4*)x)[i];
  v.x *= inv; v.y *= inv; v.z *= inv; v.w *= inv;
  ((float4*)x)[i] = v;
}

// ---------------------------------------------------------------- launcher
extern "C" void kernel_launch(void* const* d_in, const int* in_sizes, int n_in,
                              void* d_out, int out_size, void* d_ws, size_t ws_size,
                              hipStream_t stream) {
  const float* feat = (const float*)d_in[0];
  const float* emb  = (const float*)d_in[1];
  const int*   ef   = (const int*)d_in[2];
  const int*   src  = (const int*)d_in[3];
  const int*   dst  = (const int*)d_in[4];

  const int ND = in_sizes[0];
  const int NT = in_sizes[1];
  const int E  = in_sizes[2];
  const int N  = ND / DFEAT;
  const int nd4 = ND / 4;

  float* x = (float*)d_out;        // rst lives here (built in place)
  float* a = x + ND;               // attention output slot, used as working a[]

  // workspace carve (256B aligned sections)
  char* w = (char*)d_ws;
  auto carve = [&](size_t bytes) { char* p = w; w += (bytes + 255) & ~(size_t)255; return p; };
  float*    aggft = (float*)carve((size_t)ND * sizeof(float));
  float*    aggas = (float*)carve((size_t)N * sizeof(float));
  float*    ssum  = (float*)carve((size_t)N * sizeof(float));
  float*    mmax  = (float*)carve((size_t)N * sizeof(float));
  unsigned* mask  = (unsigned*)carve((size_t)N * sizeof(unsigned));
  float*    attn  = (float*)carve((size_t)N * sizeof(float));

  const int T = 256;
  const int gN  = (N + T - 1) / T;
  const int gE  = (E + T - 1) / T;
  const int gV  = (nd4 + T - 1) / T;
  const int gAg = 1024;            // persistent-ish blocks; chunk loop inside

  // --- edge softmax ---
  k_init_nodes<<<gN, T, 0, stream>>>(ssum, mask, attn, N);
  k_init_x<<<gV, T, 0, stream>>>(x, feat, nd4);
  k_edge_mask<<<gE, T, 0, stream>>>(ef, dst, mask, E);
  k_node_max<<<gN, T, 0, stream>>>(mask, emb, mmax, N, NT);
  k_edge_exp<<<gE, T, 0, stream>>>(ef, dst, emb, mmax, a, ssum, E);
  k_edge_norm<<<gE, T, 0, stream>>>(dst, ssum, a, E);

  // --- ORDER rounds of attention-weighted propagation ---
  for (int it = 0; it < ORDER; ++it) {
    k_zero<<<gV, T, 0, stream>>>(aggft, aggas, nd4, N);
    k_edge_agg<<<gAg, T, 0, stream>>>(src, dst, a, x, attn, aggft, aggas, E);
    k_node_update<<<gV, T, 0, stream>>>(feat, aggft, aggas, x, attn, nd4, N);
  }

  // --- rst = x / attn_sum ---
  k_finalize<<<gV, T, 0, stream>>>(x, attn, nd4);
}